// Encoder_Baseline_29987461660940
// MI455X (gfx1250) — compile-verified
//
#include <hip/hip_runtime.h>
#include <hip/hip_bf16.h>

typedef __attribute__((ext_vector_type(2))) float v2f;
typedef __attribute__((ext_vector_type(8))) float v8f;
typedef __attribute__((ext_vector_type(4))) unsigned int v4u;
typedef __attribute__((ext_vector_type(8))) int v8i;
typedef __attribute__((ext_vector_type(4))) int v4i;

constexpr int H  = 512;    // hidden
constexpr int IS = 2048;   // input size (stream 1)
constexpr int FS = 1536;   // feats size (stream 0)
constexpr int NB = 128;    // batch
constexpr int NT = 64;     // timesteps
constexpr int MR = NB * NT; // 8192 rows for the batched GEMMs

// ---------------- GEMM: OUT[M,N] = X[M,K] * W[N,K]^T + bias[N] ----------------
// 256 threads = 8 waves. Block tile 128x128. Wave grid 4(M) x 2(N):
// wave tile 32x64 -> 2 (M) x 4 (N) f32 16x16 accumulators.
// Operand staging: Tensor Data Mover (one 128x32 tile per tensor_load_to_lds),
// double-buffered; TDM pads 2 DWORDs per 32-DWORD row -> LDS row stride 34 floats
// (136 B: 8B-aligned for ds_load_b64 fragments, decent bank spread).
constexpr int BM = 128;
constexpr int BN = 128;
constexpr int KT = 32;
constexpr int LDSP = KT + 2;   // 34 floats effective row stride (TDM pad)

__device__ __forceinline__ float sigmoidf_(float x) {
    return 1.0f / (1.0f + __expf(-x));
}

// Issue one TDM load of a 2D tile: KT columns x BM rows of f32 from row-major
// [*, Kdim] matrix starting at gptr, into LDS at byte offset lds_off.
__device__ __forceinline__ void tdm_load_tile(const float* gptr, unsigned lds_off,
                                              int Kdim) {
    const unsigned long long ga = (unsigned long long)(uintptr_t)gptr;
    v4u g0;
    g0[0] = 1u;                                   // count=1 (valid user descriptor)
    g0[1] = lds_off;                              // lds_addr (bytes)
    g0[2] = (unsigned)(ga & 0xFFFFFFFFu);         // global_addr[31:0]
    g0[3] = (unsigned)((ga >> 32) & 0x01FFFFFFu)  // global_addr[56:32]
          | (2u << 30);                           // type=2 ("image")
    v8i g1;
    // data_size=2 (4B) | pad_enable | pad_interval=4 (32 DW) | pad_amount=1 (2 DW)
    g1[0] = (int)((2u << 16) | (1u << 20) | (4u << 22) | (1u << 25));
    // tensor_dim0 = Kdim at bits[79:48]; tensor_dim1 = 2^20 at bits[111:80]
    const unsigned kd = (unsigned)Kdim;
    const unsigned rows = 1u << 20;
    g1[1] = (int)((kd & 0xFFFFu) << 16);
    g1[2] = (int)(((kd >> 16) & 0xFFFFu) | ((rows & 0xFFFFu) << 16));
    g1[3] = (int)(((rows >> 16) & 0xFFFFu) | ((unsigned)KT << 16)); // tile_dim0=KT
    g1[4] = (int)(unsigned)BM;                    // tile_dim1=128, tile_dim2=0
    g1[5] = Kdim;                                 // tensor_dim0_stride (elements)
    g1[6] = 0;
    g1[7] = 0;
    const v4i z4 = {};                            // 2D tensor: groups 2/3 disabled
    const v8i z8 = {};
    __builtin_amdgcn_tensor_load_to_lds(g0, g1, z4, z4, z8, 0);
}

__global__ __launch_bounds__(256)
void gemm_bias_wmma(const float* __restrict__ X, const float* __restrict__ W,
                    const float* __restrict__ bias, float* __restrict__ OUT,
                    int Ndim, int Kdim)
{
    __shared__ float Xs[2][BM * LDSP];
    __shared__ float Ws[2][BN * LDSP];

    const int tid  = threadIdx.x;
    const int lane = tid & 31;
    const int wave = tid >> 5;
    const int wm   = wave & 3;        // 4 waves along M
    const int wn   = wave >> 2;       // 2 waves along N
    const int l15  = lane & 15;
    const int half = lane >> 4;       // 0: K=k,k+1   1: K=k+2,k+3

    const int m0b = blockIdx.y * BM;
    const int n0b = blockIdx.x * BN;
    const int wm0 = wm * 32;
    const int wn0 = wn * 64;

    v8f acc[2][4];
    #pragma unroll
    for (int i = 0; i < 2; ++i)
        #pragma unroll
        for (int j = 0; j < 4; ++j) acc[i][j] = (v8f){};

    const int nst = Kdim / KT;

    // Prologue: stage 0 into buffer 0 (wave 0 drives the TDM).
    if (wave == 0) {
        tdm_load_tile(X + (size_t)m0b * Kdim, (unsigned)(uintptr_t)&Xs[0][0], Kdim);
        tdm_load_tile(W + (size_t)n0b * Kdim, (unsigned)(uintptr_t)&Ws[0][0], Kdim);
    }

    for (int s = 0; s < nst; ++s) {
        const int cur = s & 1;
        if (wave == 0) __builtin_amdgcn_s_wait_tensorcnt(0);  // stage s landed
        __syncthreads();  // everyone sees buf[cur]; buf[cur^1] readers (s-1) done
        if (wave == 0 && s + 1 < nst) {
            const int k1 = (s + 1) * KT;   // overlap: DMA stage s+1 during compute
            tdm_load_tile(X + (size_t)m0b * Kdim + k1,
                          (unsigned)(uintptr_t)&Xs[cur ^ 1][0], Kdim);
            tdm_load_tile(W + (size_t)n0b * Kdim + k1,
                          (unsigned)(uintptr_t)&Ws[cur ^ 1][0], Kdim);
        }

        const float* xs = &Xs[cur][0];
        const float* ws = &Ws[cur][0];
        #pragma unroll
        for (int kk = 0; kk < KT; kk += 4) {
            const int koff = kk + 2 * half;
            v2f a[2], b[4];
            #pragma unroll
            for (int i = 0; i < 2; ++i)
                a[i] = *(const v2f*)&xs[(wm0 + i * 16 + l15) * LDSP + koff];
            #pragma unroll
            for (int j = 0; j < 4; ++j)
                b[j] = *(const v2f*)&ws[(wn0 + j * 16 + l15) * LDSP + koff];
            #pragma unroll
            for (int i = 0; i < 2; ++i)
                #pragma unroll
                for (int j = 0; j < 4; ++j)
                    acc[i][j] = __builtin_amdgcn_wmma_f32_16x16x4_f32(
                        false, a[i], false, b[j], (short)0, acc[i][j], false, false);
        }
    }

    // Epilogue: bias + store (C layout: VGPR r -> M=m0+r (lanes 0-15) / m0+8+r)
    #pragma unroll
    for (int i = 0; i < 2; ++i) {
        #pragma unroll
        for (int j = 0; j < 4; ++j) {
            const int n = n0b + wn0 + j * 16 + l15;
            const float bv = bias[n];
            #pragma unroll
            for (int rr = 0; rr < 8; ++rr) {
                const int m = m0b + wm0 + i * 16 + rr + 8 * half;
                OUT[(size_t)m * Ndim + n] = acc[i][j][rr] + bv;
            }
        }
    }
}

// ---------------- GRU recurrent step (one launch per timestep) ----------------
// 256 threads = 8 waves. Block tile: M=64 (batch) x N=64 (hidden cols).
// Wave tile 16x32 per gate; 3 gates (r,i,n) -> acc[3][2] 16x16 f32 tiles.
// gh fragments come straight from global (W_hh is 3 MB -> L2 resident).
__global__ __launch_bounds__(256)
void gru_step_wmma(const float* __restrict__ gi, const float* __restrict__ gf,
                   const float* __restrict__ Whh, const float* __restrict__ b_hh,
                   const float* __restrict__ h_prev, float* __restrict__ h_next,
                   float* __restrict__ eo, int t)
{
    const int tid  = threadIdx.x;
    const int lane = tid & 31;
    const int wave = tid >> 5;
    const int wm   = wave & 3;        // 4 waves along M (batch)
    const int wn   = wave >> 2;       // 2 waves along N (hidden)
    const int l15  = lane & 15;
    const int half = lane >> 4;

    const int m0 = blockIdx.y * 64 + wm * 16;   // batch row base
    const int n0 = blockIdx.x * 64 + wn * 32;   // hidden col base

    v8f acc[3][2];
    #pragma unroll
    for (int g = 0; g < 3; ++g)
        #pragma unroll
        for (int j = 0; j < 2; ++j) acc[g][j] = (v8f){};

    #pragma unroll 4
    for (int k = 0; k < H; k += 4) {
        const int koff = k + 2 * half;
        const v2f a = *(const v2f*)(h_prev + (size_t)(m0 + l15) * H + koff);
        #pragma unroll
        for (int g = 0; g < 3; ++g) {
            #pragma unroll
            for (int j = 0; j < 2; ++j) {
                const v2f b = *(const v2f*)(Whh + (size_t)(g * H + n0 + j * 16 + l15) * H + koff);
                acc[g][j] = __builtin_amdgcn_wmma_f32_16x16x4_f32(
                    false, a, false, b, (short)0, acc[g][j], false, false);
            }
        }
    }

    // Fused gate epilogue.
    #pragma unroll
    for (int j = 0; j < 2; ++j) {
        const int col = n0 + j * 16 + l15;
        const float bh_r = b_hh[col];
        const float bh_i = b_hh[H + col];
        const float bh_n = b_hh[2 * H + col];
        #pragma unroll
        for (int rr = 0; rr < 8; ++rr) {
            const int bidx = m0 + rr + 8 * half;
            const size_t bt = (size_t)bidx * NT + t;
            const size_t i3 = bt * (3 * H);
            const size_t i2 = bt * (2 * H);
            const float i_r = gi[i3 + col];
            const float i_i = gi[i3 + H + col];
            const float i_n = gi[i3 + 2 * H + col];
            const float f_r = gf[i2 + col];
            const float f_i = gf[i2 + H + col];
            const float h_r = acc[0][j][rr] + bh_r;
            const float h_i = acc[1][j][rr] + bh_i;
            const float h_n = acc[2][j][rr] + bh_n;
            const float rg = sigmoidf_(i_r + h_r + f_r);
            const float ig = sigmoidf_(i_i + h_i + f_i);
            const float ng = tanhf(i_n + rg * h_n);
            const float hold = h_prev[(size_t)bidx * H + col];
            const float hy = ng + ig * (hold - ng);
            h_next[(size_t)bidx * H + col] = hy;
            eo[bt * H + col] = hy;
        }
    }
}

__global__ void zero_kernel(float* __restrict__ p, int n) {
    const int i = blockIdx.x * blockDim.x + threadIdx.x;
    if (i < n) p[i] = 0.0f;
}

extern "C" void kernel_launch(void* const* d_in, const int* in_sizes, int n_in,
                              void* d_out, int out_size, void* d_ws, size_t ws_size,
                              hipStream_t stream) {
    const float* feat0 = (const float*)d_in[0];   // [B,T,FS]
    const float* feat1 = (const float*)d_in[1];   // [B,T,IS]
    const float* W_ih  = (const float*)d_in[2];   // [3H,IS]
    const float* b_ih  = (const float*)d_in[3];   // [3H]
    const float* W_hh  = (const float*)d_in[4];   // [3H,H]
    const float* b_hh  = (const float*)d_in[5];   // [3H]
    const float* W_fh  = (const float*)d_in[6];   // [2H,FS]
    const float* b_fh  = (const float*)d_in[7];   // [2H]

    float* eo = (float*)d_out;                    // [B,T,H]
    float* hT = eo + (size_t)NB * NT * H;         // [B,H]

    float* ws = (float*)d_ws;
    float* gi = ws;                               // [MR, 3H]
    float* gf = gi + (size_t)MR * 3 * H;          // [MR, 2H]
    float* h0 = gf + (size_t)MR * 2 * H;          // [NB, H] ping
    float* h1 = h0 + (size_t)NB * H;              // [NB, H] pong

    // h(0) = 0
    zero_kernel<<<(NB * H + 255) / 256, 256, 0, stream>>>(h0, NB * H);

    // Batched input / auxiliary projections (fp32 WMMA GEMMs, TDM-staged).
    gemm_bias_wmma<<<dim3(3 * H / BN, MR / BM), 256, 0, stream>>>(
        feat1, W_ih, b_ih, gi, 3 * H, IS);
    gemm_bias_wmma<<<dim3(2 * H / BN, MR / BM), 256, 0, stream>>>(
        feat0, W_fh, b_fh, gf, 2 * H, FS);

    // Sequential recurrence: stream ordering carries the dependency.
    float* hp = h0;
    float* hn = h1;
    for (int t = 0; t < NT; ++t) {
        gru_step_wmma<<<dim3(H / 64, NB / 64), 256, 0, stream>>>(
            gi, gf, W_hh, b_hh, hp, hn, eo, t);
        float* tmp = hp; hp = hn; hn = tmp;
    }

    // Final hidden state -> second output.
    (void)hipMemcpyAsync(hT, hp, (size_t)NB * H * sizeof(float),
                         hipMemcpyDeviceToDevice, stream);
}